// CaptionModel_45251775431013
// MI455X (gfx1250) — compile-verified
//
#include <hip/hip_runtime.h>
#include <hip/hip_bf16.h>

// Problem constants from the reference: B=32, K=5, T=20, V=9488, H=1024
#define BB 32
#define KK 5
#define TT 20
#define VV 9488
#define HH 1024
#define V4 (VV / 4)               // 2372 float4 per row
#define NBLK (V4 / 256)           // 9 full 256-lane float4 blocks
#define FULL4 (NBLK * 256)        // 2304

typedef unsigned int u32;
typedef u32 u32x4 __attribute__((ext_vector_type(4)));
typedef int  i32x4 __attribute__((ext_vector_type(4)));
typedef int  i32x8 __attribute__((ext_vector_type(8)));

// ---------- register-resident sorted top-5 insert (fully unrolled -> VGPRs) ----------
__device__ __forceinline__ void ins5(float (&v)[5], int (&id)[5], float nv, int nid) {
  if (nv <= v[4]) return;
  v[4] = nv; id[4] = nid;
#pragma unroll
  for (int j = 4; j > 0; --j) {
    if (v[j] > v[j - 1]) {
      float tv = v[j]; v[j] = v[j - 1]; v[j - 1] = tv;
      int ti = id[j]; id[j] = id[j - 1]; id[j - 1] = ti;
    }
  }
}

// =======================================================================
// Kernel A: per-batch top-5 over K*V candidates; emits beam_ix to ws,
// writes beam_seq (out0) and beam_logprobs_sum (out2) outputs.
// =======================================================================
__global__ void __launch_bounds__(256)
topk_kernel(const float* __restrict__ logprobs,   // (B*K, V)
            const int*   __restrict__ beam_seq,   // (B, K, T)
            const float* __restrict__ blps,       // (B, K)
            float* __restrict__ out0,             // (B, K, T+1) as float
            float* __restrict__ out2,             // (B, K)
            int*   __restrict__ ws_bix)           // (B*K)
{
  const int b   = blockIdx.x;   // 0..31
  const int tid = threadIdx.x;  // 0..255

  float v[5];  int id[5];
#pragma unroll
  for (int j = 0; j < 5; ++j) { v[j] = -INFINITY; id[j] = 0; }

#pragma unroll 1
  for (int k = 0; k < KK; ++k) {
    const float s = blps[b * KK + k];
    const float4* lp4 = (const float4*)(logprobs + (size_t)(b * KK + k) * VV);
    const int fbase = k * VV;
#pragma unroll 1
    for (int i = tid; i < V4; i += 256) {
      float4 x = lp4[i];
      const int f = fbase + i * 4;
      ins5(v, id, s + x.x, f + 0);
      ins5(v, id, s + x.y, f + 1);
      ins5(v, id, s + x.z, f + 2);
      ins5(v, id, s + x.w, f + 3);
    }
  }

  // LDS tree-merge of 256 sorted-5 lists
  __shared__ float sv[256 * 5];
  __shared__ int   si[256 * 5];
#pragma unroll
  for (int j = 0; j < 5; ++j) { sv[tid * 5 + j] = v[j]; si[tid * 5 + j] = id[j]; }
  __syncthreads();
  for (int s = 128; s > 0; s >>= 1) {
    if (tid < s) {
      const int o = (tid + s) * 5;
#pragma unroll
      for (int j = 0; j < 5; ++j) ins5(v, id, sv[o + j], si[o + j]);
#pragma unroll
      for (int j = 0; j < 5; ++j) { sv[tid * 5 + j] = v[j]; si[tid * 5 + j] = id[j]; }
    }
    __syncthreads();
  }

  if (tid == 0) {
#pragma unroll 1
    for (int k = 0; k < KK; ++k) {
      const int ix  = id[k];
      const int bix = ix / VV;
      const int sel = ix - bix * VV;
      ws_bix[b * KK + k] = bix;
      out2[b * KK + k]   = v[k];   // ys == gathered sum + gathered lp
      const int* srcseq = beam_seq + (size_t)(b * KK + bix) * TT;  // t>0 path
      float* dstseq     = out0 + (size_t)(b * KK + k) * (TT + 1);
#pragma unroll 1
      for (int t = 0; t < TT; ++t) dstseq[t] = (float)srcseq[t];
      dstseq[TT] = (float)sel;
    }
  }
}

// =======================================================================
// Kernel B: dominant gather-copy, async global->LDS->global pipeline with
// 9-deep MLP: issue all async loads into 9 LDS slots, one asynccnt wait,
// then all async stores. IOFFSET applies to both LDS and global addresses
// and both strides are 4096B/iter, so one base VGPR pair + immediate
// offsets cover the whole row.
// =======================================================================
__global__ void __launch_bounds__(256)
gather_lp_kernel(const float* __restrict__ logprobs,  // (B*K, V)
                 const float* __restrict__ bslp,      // (B, K, T, V)
                 const int*   __restrict__ ws_bix,    // (B*K)
                 float* __restrict__ out1)            // (B, K, T+1, V)
{
  const int row = blockIdx.x;            // 0 .. 3359
  const int tid = threadIdx.x;
  const int b   = row / (KK * (TT + 1));
  const int rem = row - b * (KK * (TT + 1));
  const int k   = rem / (TT + 1);
  const int t   = rem - k * (TT + 1);
  const int bix = ws_bix[b * KK + k];

  const float* src = (t < TT)
      ? (bslp + ((size_t)((b * KK + bix) * TT + t)) * VV)
      : (logprobs + (size_t)(b * KK + bix) * VV);
  float* dst = out1 + (size_t)row * VV;

  const float4* s4 = (const float4*)src + tid;   // lane base, stride 4096B/iter
  float4*       d4 = (float4*)dst + tid;

  __shared__ float4 stage[NBLK][256];            // 9 * 4KB = 36KB staging
  const unsigned int lds = (unsigned int)(unsigned long long)(&stage[0][tid]);

  // phase 1: 9 async loads in flight
#pragma unroll
  for (int it = 0; it < NBLK; ++it) {
    asm volatile("global_load_async_to_lds_b128 %0, %1, off offset:%2"
                 :: "v"(lds), "v"(s4), "i"(it * 4096) : "memory");
  }
  asm volatile("s_wait_asynccnt 0x0" ::: "memory");
  // phase 2: 9 async stores in flight (S_ENDPGM waits all counters)
#pragma unroll
  for (int it = 0; it < NBLK; ++it) {
    asm volatile("global_store_async_from_lds_b128 %0, %1, off offset:%2"
                 :: "v"(d4), "v"(lds), "i"(it * 4096) : "memory");
  }

  // tail: 2372 - 2304 = 68 float4, direct b128 copy
  const int i = FULL4 + tid - tid + (FULL4 + tid < V4 ? 0 : 0); // keep simple below
  if (FULL4 + tid < V4) {
    ((float4*)dst)[FULL4 + tid] = ((const float4*)src)[FULL4 + tid];
  }
  (void)i;
}

// =======================================================================
// Kernel C: state gather via the Tensor Data Mover. One wave per block;
// D# describes a 1D 1024-element f32 tile. TDM ignores EXEC, so exactly
// one tensor_load_to_lds / tensor_store_from_lds pair issues per row.
// =======================================================================
__global__ void __launch_bounds__(32)
state_kernel(const float* __restrict__ state,   // (2, B*K, H)
             const int*   __restrict__ ws_bix,  // (B*K)
             float* __restrict__ out3)          // (2, B*K, H)
{
  __shared__ float rowbuf[HH];                  // 4KB staging
  const int j     = blockIdx.x;                 // 0..319
  const int layer = j / (BB * KK);
  const int idx   = j - layer * (BB * KK);
  const int b     = idx / KK;
  const int bix   = ws_bix[idx];

  const unsigned long long src =
      (unsigned long long)(const void*)(state + ((size_t)layer * (BB * KK) + (size_t)b * KK + bix) * HH);
  const unsigned long long dst =
      (unsigned long long)(void*)(out3 + (size_t)j * HH);
  const u32 lds = (u32)(unsigned long long)(&rowbuf[0]);

  // D# group1: data_size=4B (code 2), tensor_dim0=1024, tensor_dim1=1,
  // tile_dim0=1024, tile_dim1=1, tensor_dim0_stride=1024
  i32x8 g1;
  g1[0] = (2 << 16);                 // wg_mask=0 | data_size=2
  g1[1] = (int)((HH & 0xffff) << 16);// tensor_dim0[15:0] at bits 63:48
  g1[2] = (HH >> 16) | (1 << 16);    // tensor_dim0[31:16] | tensor_dim1[15:0]=1
  g1[3] = (int)((u32)HH << 16);      // tile_dim0=1024 at bits 127:112
  g1[4] = 1;                         // tile_dim1=1, tile_dim2=0
  g1[5] = HH;                        // tensor_dim0_stride[31:0]
  g1[6] = 0;
  g1[7] = 0;
  i32x4 gz = {0, 0, 0, 0};

  u32x4 g0;
  g0[0] = 1u;                                            // count=1
  g0[1] = lds;                                           // lds_addr
  g0[2] = (u32)src;                                      // global_addr[31:0]
  g0[3] = (u32)((src >> 32) & 0x01ffffffu) | (2u << 30); // addr[56:32] | type=2

#if defined(__clang_major__) && __clang_major__ >= 23
  i32x8 gz8 = {0, 0, 0, 0, 0, 0, 0, 0};
  __builtin_amdgcn_tensor_load_to_lds(g0, g1, gz, gz, gz8, 0);
  __builtin_amdgcn_s_wait_tensorcnt(0);
  g0[2] = (u32)dst;
  g0[3] = (u32)((dst >> 32) & 0x01ffffffu) | (2u << 30);
  __builtin_amdgcn_tensor_store_from_lds(g0, g1, gz, gz, gz8, 0);
#else
  __builtin_amdgcn_tensor_load_to_lds(g0, g1, gz, gz, 0);
  __builtin_amdgcn_s_wait_tensorcnt(0);
  g0[2] = (u32)dst;
  g0[3] = (u32)((dst >> 32) & 0x01ffffffu) | (2u << 30);
  __builtin_amdgcn_tensor_store_from_lds(g0, g1, gz, gz, 0);
#endif
  __builtin_amdgcn_s_wait_tensorcnt(0);
}

extern "C" void kernel_launch(void* const* d_in, const int* in_sizes, int n_in,
                              void* d_out, int out_size, void* d_ws, size_t ws_size,
                              hipStream_t stream) {
  const float* logprobs = (const float*)d_in[0];   // (B*K, V) f32
  const int*   beam_seq = (const int*)  d_in[1];   // (B, K, T) i32
  const float* bslp     = (const float*)d_in[2];   // (B, K, T, V) f32
  const float* blps     = (const float*)d_in[3];   // (B, K) f32
  const float* state    = (const float*)d_in[4];   // (2, B*K, H) f32
  // d_in[5] = beam_size (5), d_in[6] = t (20): constants, hardcoded.

  float* out  = (float*)d_out;
  float* out0 = out;                                          // beam_seq: 3360
  float* out1 = out + (size_t)BB * KK * (TT + 1);             // beam_seq_logprobs: 31,879,680
  float* out2 = out1 + (size_t)BB * KK * (TT + 1) * VV;       // beam_logprobs_sum: 160
  float* out3 = out2 + (size_t)BB * KK;                       // new_state: 327,680

  int* ws_bix = (int*)d_ws;                                   // 160 ints

  topk_kernel<<<BB, 256, 0, stream>>>(logprobs, beam_seq, blps, out0, out2, ws_bix);
  gather_lp_kernel<<<BB * KK * (TT + 1), 256, 0, stream>>>(logprobs, bslp, ws_bix, out1);
  state_kernel<<<2 * BB * KK, 32, 0, stream>>>(state, ws_bix, out3);
}